// SWD16_28449863369560
// MI455X (gfx1250) — compile-verified
//
#include <hip/hip_runtime.h>
#include <float.h>

// Problem constants (match the reference)
#define NB   8
#define NL   6656
#define ND   1024
#define NW   13      // window (sort group size)
#define NG   512     // NL / NW
#define ROLL 5

// CDNA5 async global->LDS path, guarded so compilation never fails.
#if defined(__gfx1250__) && __has_builtin(__builtin_amdgcn_global_load_async_to_lds_b128)
#define USE_ASYNC_LDS 1
#endif

#if defined(USE_ASYNC_LDS)
// Builtin signature is (v4i addrspace(1)*, v4i addrspace(3)*, imm offset, imm cpol)
typedef int v4i __attribute__((ext_vector_type(4)));
typedef v4i __attribute__((address_space(1)))* gv4i_p;
typedef v4i __attribute__((address_space(3)))* lv4i_p;
#endif

__device__ __forceinline__ void ce(float4 &a, float4 &b) {
    float t;
    t = fminf(a.x, b.x); b.x = fmaxf(a.x, b.x); a.x = t;
    t = fminf(a.y, b.y); b.y = fmaxf(a.y, b.y); a.y = t;
    t = fminf(a.z, b.z); b.z = fmaxf(a.z, b.z); a.z = t;
    t = fminf(a.w, b.w); b.w = fmaxf(a.w, b.w); a.w = t;
}

// One block per (b, j): 256 threads x float4 covers the full D=1024 row width.
// Thread gathers 13 strided rows, sorts 13 values per lane-column via a
// bitonic-16 network (padded with +FLT_MAX), scatters ranks back.
__global__ __launch_bounds__(256) void sort13_kernel(const float* __restrict__ v,
                                                     float* __restrict__ out) {
    const int t = threadIdx.x;            // 0..255 -> d-chunk
    const int b = blockIdx.x >> 9;        // / 512
    const int j = blockIdx.x & (NG - 1);  // % 512
    const long base = (long)b * (long)(NL * ND);
    const int d = t << 2;

    int rows[NW];
#pragma unroll
    for (int w = 0; w < NW; ++w) {
        int r = w * NG + j + ROLL;        // max 6660 < 2*NL -> one conditional wrap
        rows[w] = (r >= NL) ? (r - NL) : r;
    }

    float4 a[16];

#if defined(USE_ASYNC_LDS)
    __shared__ float tile[NW * ND];       // 13 * 4KB = 52KB (<< 320KB/WGP)
#pragma unroll
    for (int w = 0; w < NW; ++w) {
        const float* gsrc = v + base + (long)rows[w] * ND + d;
        float* ldst = tile + w * ND + d;
        __builtin_amdgcn_global_load_async_to_lds_b128(
            (gv4i_p)gsrc, (lv4i_p)ldst, 0, 0);
    }
#if __has_builtin(__builtin_amdgcn_s_wait_asynccnt)
    __builtin_amdgcn_s_wait_asynccnt(0);
#else
    asm volatile("s_wait_asynccnt 0" ::: "memory");
#endif
    __syncthreads();
#pragma unroll
    for (int w = 0; w < NW; ++w)
        a[w] = *(const float4*)(tile + w * ND + d);
#else
    // Fallback: direct coalesced global float4 loads.
#pragma unroll
    for (int w = 0; w < NW; ++w)
        a[w] = *(const float4*)(v + base + (long)rows[w] * ND + d);
#endif

#pragma unroll
    for (int w = NW; w < 16; ++w)
        a[w] = make_float4(FLT_MAX, FLT_MAX, FLT_MAX, FLT_MAX);

    // Fully unrolled ascending bitonic sort of 16 register float4s.
#pragma unroll
    for (int k = 2; k <= 16; k <<= 1) {
#pragma unroll
        for (int s = k >> 1; s > 0; s >>= 1) {
#pragma unroll
            for (int i = 0; i < 16; ++i) {
                const int l = i ^ s;
                if (l > i) {
                    if ((i & k) == 0) ce(a[i], a[l]);
                    else              ce(a[l], a[i]);
                }
            }
        }
    }

#pragma unroll
    for (int w = 0; w < NW; ++w)
        *(float4*)(out + base + (long)rows[w] * ND + d) = a[w];
}

extern "C" void kernel_launch(void* const* d_in, const int* in_sizes, int n_in,
                              void* d_out, int out_size, void* d_ws, size_t ws_size,
                              hipStream_t stream) {
    // setup_inputs order: q, k, v. Only v is used.
    const float* v = (const float*)d_in[2];
    float* out = (float*)d_out;

    dim3 grid(NB * NG);   // 4096 blocks: one per (b, j)
    dim3 block(256);      // 8 wave32 waves; t*4 spans D=1024 exactly
    sort13_kernel<<<grid, block, 0, stream>>>(v, out);
}